// GAT_12816182411599
// MI455X (gfx1250) — compile-verified
//
#include <hip/hip_runtime.h>
#include <hip/hip_bf16.h>

// ---------------------------------------------------------------------------
// GAT (3x GATConv + add-pool + MLP) for MI455X (gfx1250, wave32, WMMA).
// Dense transforms use V_WMMA_F32_16X16X4_F32 (full fp32 matrix path).
// Edge softmax/aggregation is bandwidth-bound -> wave-per-(edge,head),
// lane == channel (C==32 fits wave32 exactly).
// ---------------------------------------------------------------------------

#define NNODE 50000
#define NEDGE 800000
#define IN_DIM 64
#define NHEAD 4
#define CH 32
#define HC 128
#define ED 16
#define NGRAPH 2000

typedef float v2f __attribute__((ext_vector_type(2)));
typedef float v8f __attribute__((ext_vector_type(8)));

// ---------------------------------------------------------------------------
// Generic fp32 WMMA GEMM: C[M x Ncols] = A[M x K] @ B[K x Ncols]
// One wave per 16x16 output tile. M % 16 == 0, K % 4 == 0, Ncols % 16 == 0.
// A fragment (16x4 f32): lane r=lane&15 holds row M=r, K pair {2*half, 2*half+1}
// B fragment (4x16 f32): lane r holds col N=r, same K pair. (ISA 7.12.2)
// ---------------------------------------------------------------------------
__global__ void gat_gemm_wmma_f32(const float* __restrict__ A,
                                  const float* __restrict__ B,
                                  float* __restrict__ C,
                                  int M, int K, int Ncols) {
    int wave = blockIdx.x * (blockDim.x >> 5) + (threadIdx.x >> 5);
    int tilesN = Ncols >> 4;
    int totalTiles = (M >> 4) * tilesN;
    if (wave >= totalTiles) return;              // whole-wave uniform exit
    int tm = wave / tilesN;
    int tn = wave - tm * tilesN;
    int lane = threadIdx.x & 31;
    int half = lane >> 4;
    int r = lane & 15;

    const float* arow = A + (size_t)(tm * 16 + r) * K;
    const float* bcol = B + tn * 16 + r;

    v8f c = {};
    for (int k = 0; k < K; k += 4) {
        int kk = k + half * 2;
        v2f a, b;
        a.x = arow[kk];
        a.y = arow[kk + 1];
        b.x = bcol[(size_t)kk * Ncols];
        b.y = bcol[(size_t)(kk + 1) * Ncols];
        c = __builtin_amdgcn_wmma_f32_16x16x4_f32(
                false, a, false, b, (short)0, c, false, false);
    }
    // C/D layout: VGPR v -> row tm*16 + v + half*8, col tn*16 + r
    float* crow = C + (size_t)(tm * 16 + half * 8) * Ncols + tn * 16 + r;
#pragma unroll
    for (int v = 0; v < 8; ++v) crow[(size_t)v * Ncols] = c[v];
}

// ---------------------------------------------------------------------------
// Fused MLP head: A row = [h1 | h2 | h3 | hpool[batch[row]]] (K=512, gathered),
// B = lw1 (512x512). Epilogue: leaky_relu(c + lb1, 0.01) * lw2[col],
// shuffle-reduce across 16 cols per half, atomicAdd into out[row] (pre-init to lb2).
// ---------------------------------------------------------------------------
__global__ void gat_mlp_fused_wmma(const float* __restrict__ h1,
                                   const float* __restrict__ h2,
                                   const float* __restrict__ h3,
                                   const float* __restrict__ hpool,
                                   const int* __restrict__ batch,
                                   const float* __restrict__ lw1,
                                   const float* __restrict__ lb1,
                                   const float* __restrict__ lw2,
                                   float* __restrict__ out, int M) {
    const int NC = 4 * HC;                       // 512
    int wave = blockIdx.x * (blockDim.x >> 5) + (threadIdx.x >> 5);
    int tilesN = NC >> 4;                        // 32
    int totalTiles = (M >> 4) * tilesN;
    if (wave >= totalTiles) return;
    int tm = wave / tilesN;
    int tn = wave - tm * tilesN;
    int lane = threadIdx.x & 31;
    int half = lane >> 4;
    int r = lane & 15;
    int row = tm * 16 + r;
    int g = batch[row];

    const float* sptr[4];
    sptr[0] = h1 + (size_t)row * HC;
    sptr[1] = h2 + (size_t)row * HC;
    sptr[2] = h3 + (size_t)row * HC;
    sptr[3] = hpool + (size_t)g * HC;

    v8f c = {};
#pragma unroll
    for (int s = 0; s < 4; ++s) {
        const float* sp = sptr[s];
        const float* bp = lw1 + (size_t)(s * HC) * NC + tn * 16 + r;
        for (int k = 0; k < HC; k += 4) {
            int kk = k + half * 2;
            v2f a, b;
            a.x = sp[kk];
            a.y = sp[kk + 1];
            b.x = bp[(size_t)kk * NC];
            b.y = bp[(size_t)(kk + 1) * NC];
            c = __builtin_amdgcn_wmma_f32_16x16x4_f32(
                    false, a, false, b, (short)0, c, false, false);
        }
    }

    int col = tn * 16 + r;
    float w2 = lw2[col];
    float bb = lb1[col];
    int baserow = tm * 16 + half * 8;
#pragma unroll
    for (int v = 0; v < 8; ++v) {
        float val = c[v] + bb;
        val = val > 0.f ? val : 0.01f * val;     // leaky_relu 0.01
        val *= w2;
        // reduce 16 cols held by this half of the wave
#pragma unroll
        for (int off = 1; off < 16; off <<= 1)
            val += __shfl_xor(val, off, 32);
        if (r == 0) atomicAdd(&out[baserow + v], val);
    }
}

// ---------------------------------------------------------------------------
// Me[l][d][h] = sum_c We_l[d, h*CH + c] * ae_l[h, c]   (3 x 16 x 4)
// ---------------------------------------------------------------------------
__global__ void gat_compute_me(const float* __restrict__ We1, const float* __restrict__ ae1,
                               const float* __restrict__ We2, const float* __restrict__ ae2,
                               const float* __restrict__ We3, const float* __restrict__ ae3,
                               float* __restrict__ Me) {
    int i = threadIdx.x;
    if (i >= 3 * ED * NHEAD) return;
    int l = i / (ED * NHEAD);
    int rem = i - l * (ED * NHEAD);
    int d = rem >> 2;
    int h = rem & 3;
    const float* We = (l == 0) ? We1 : (l == 1) ? We2 : We3;
    const float* ae = (l == 0) ? ae1 : (l == 1) ? ae2 : ae3;
    float s = 0.f;
#pragma unroll
    for (int c = 0; c < CH; ++c) s += We[d * HC + h * CH + c] * ae[h * CH + c];
    Me[i] = s;
}

// alpha_edge for all 3 layers in a single pass over edge_attr.
__global__ void gat_edge_feat_alpha(const float* __restrict__ edge_attr,
                                    const float* __restrict__ Me,
                                    float* __restrict__ ae_all, int E_) {
    __shared__ float sMe[3 * ED * NHEAD];
    if (threadIdx.x < 3 * ED * NHEAD) sMe[threadIdx.x] = Me[threadIdx.x];
    __syncthreads();
    int e = blockIdx.x * blockDim.x + threadIdx.x;
    if (e >= E_) return;
    float ea[ED];
#pragma unroll
    for (int d = 0; d < ED; ++d) ea[d] = edge_attr[(size_t)e * ED + d];
#pragma unroll
    for (int l = 0; l < 3; ++l) {
#pragma unroll
        for (int h = 0; h < NHEAD; ++h) {
            float s = 0.f;
#pragma unroll
            for (int d = 0; d < ED; ++d) s += ea[d] * sMe[l * 64 + d * 4 + h];
            ae_all[(size_t)l * (E_ * NHEAD) + (size_t)e * NHEAD + h] = s;
        }
    }
}

// alpha_src / alpha_dst per (node, head)
__global__ void gat_node_alpha(const float* __restrict__ h,
                               const float* __restrict__ a_s,
                               const float* __restrict__ a_d,
                               float* __restrict__ asn, float* __restrict__ adn, int NH) {
    int i = blockIdx.x * blockDim.x + threadIdx.x;
    if (i >= NH) return;
    int n = i >> 2;
    int hd = i & 3;
    const float* hp = h + (size_t)n * HC + hd * CH;
    const float* ap = a_s + hd * CH;
    const float* dp = a_d + hd * CH;
    float s1 = 0.f, s2 = 0.f;
#pragma unroll
    for (int c = 0; c < CH; ++c) {
        float v = hp[c];
        s1 += v * ap[c];
        s2 += v * dp[c];
    }
    asn[i] = s1;
    adn[i] = s2;
}

// per-layer accumulator init: hout = 0, denom = 0, amax = map(-FLT_MAX)
__global__ void gat_init_layer(float* __restrict__ hout, float* __restrict__ denom,
                               unsigned* __restrict__ amax, int NHC, int NH) {
    int i = blockIdx.x * blockDim.x + threadIdx.x;
    if (i < NHC) hout[i] = 0.f;
    if (i < NH) {
        denom[i] = 0.f;
        amax[i] = ~__float_as_uint(-3.402823466e38f);  // monotone map of -FLT_MAX
    }
}

__device__ __forceinline__ unsigned fmap(float f) {
    unsigned u = __float_as_uint(f);
    return (u & 0x80000000u) ? ~u : (u | 0x80000000u);
}
__device__ __forceinline__ float funmap(unsigned u) {
    return (u & 0x80000000u) ? __uint_as_float(u & 0x7fffffffu)
                             : __uint_as_float(~u);
}

// pass 1: alpha = leaky(as[src]+ad[dst]+ae, 0.2); segment-max via mapped atomicMax
__global__ void gat_edge_alpha(const float* __restrict__ asn, const float* __restrict__ adn,
                               const float* __restrict__ ae,
                               const int* __restrict__ src, const int* __restrict__ dst,
                               float* __restrict__ alpha, unsigned* __restrict__ amax, int EH) {
    int i = blockIdx.x * blockDim.x + threadIdx.x;
    if (i >= EH) return;
    int e = i >> 2;
    int hd = i & 3;
    float a = asn[src[e] * NHEAD + hd] + adn[dst[e] * NHEAD + hd] + ae[i];
    a = a > 0.f ? a : 0.2f * a;
    alpha[i] = a;
    atomicMax(&amax[dst[e] * NHEAD + hd], fmap(a));
}

// pass 2: ex = exp(alpha - amax[dst]); denom[dst] += ex  (ex overwrites alpha)
__global__ void gat_edge_exp(float* __restrict__ alpha_ex,
                             const unsigned* __restrict__ amax,
                             const int* __restrict__ dst,
                             float* __restrict__ denom, int EH) {
    int i = blockIdx.x * blockDim.x + threadIdx.x;
    if (i >= EH) return;
    int e = i >> 2;
    int hd = i & 3;
    float am = funmap(amax[dst[e] * NHEAD + hd]);
    float ex = __expf(alpha_ex[i] - am);
    alpha_ex[i] = ex;
    atomicAdd(&denom[dst[e] * NHEAD + hd], ex);
}

// pass 3: one wave per (edge, head), lane == channel. Coalesced gather+scatter.
__global__ void gat_edge_aggregate(const float* __restrict__ htmp,
                                   const float* __restrict__ ex,
                                   const float* __restrict__ denom,
                                   const int* __restrict__ src, const int* __restrict__ dst,
                                   float* __restrict__ hout, int EH) {
    int wid = blockIdx.x * (blockDim.x >> 5) + (threadIdx.x >> 5);
    if (wid >= EH) return;
    int lane = threadIdx.x & 31;
    int e = wid >> 2;
    int hd = wid & 3;
    int s = src[e];
    int d = dst[e];
    float w = ex[wid] / (denom[d * NHEAD + hd] + 1e-16f);
    float val = htmp[(size_t)s * HC + hd * CH + lane] * w;
    atomicAdd(&hout[(size_t)d * HC + hd * CH + lane], val);
}

__global__ void gat_add_bias(float* __restrict__ hout, const float* __restrict__ b, int NHC) {
    int i = blockIdx.x * blockDim.x + threadIdx.x;
    if (i < NHC) hout[i] += b[i & (HC - 1)];
}

// final init: hpool = 0, out = lb2
__global__ void gat_init_final(float* __restrict__ hpool, float* __restrict__ out,
                               const float* __restrict__ lb2, int GHC, int Nn) {
    int i = blockIdx.x * blockDim.x + threadIdx.x;
    if (i < GHC) hpool[i] = 0.f;
    if (i < Nn) out[i] = lb2[0];
}

// global add pool: hpool[batch[n]] += h3[n]
__global__ void gat_pool(const float* __restrict__ h3, const int* __restrict__ batch,
                         float* __restrict__ hpool, int NHC) {
    int i = blockIdx.x * blockDim.x + threadIdx.x;
    if (i >= NHC) return;
    int n = i >> 7;
    atomicAdd(&hpool[(size_t)batch[n] * HC + (i & (HC - 1))], h3[i]);
}

// ---------------------------------------------------------------------------

extern "C" void kernel_launch(void* const* d_in, const int* in_sizes, int n_in,
                              void* d_out, int out_size, void* d_ws, size_t ws_size,
                              hipStream_t stream) {
    (void)in_sizes; (void)n_in; (void)out_size; (void)ws_size;

    const float* x         = (const float*)d_in[0];
    const int*   edge_idx  = (const int*)d_in[1];
    const float* edge_attr = (const float*)d_in[2];
    const int*   batch     = (const int*)d_in[3];
    const int*   src = edge_idx;
    const int*   dst = edge_idx + NEDGE;

    const float* W[3]  = {(const float*)d_in[4],  (const float*)d_in[10], (const float*)d_in[16]};
    const float* AS[3] = {(const float*)d_in[5],  (const float*)d_in[11], (const float*)d_in[17]};
    const float* AD[3] = {(const float*)d_in[6],  (const float*)d_in[12], (const float*)d_in[18]};
    const float* WE[3] = {(const float*)d_in[7],  (const float*)d_in[13], (const float*)d_in[19]};
    const float* AE[3] = {(const float*)d_in[8],  (const float*)d_in[14], (const float*)d_in[20]};
    const float* B[3]  = {(const float*)d_in[9],  (const float*)d_in[15], (const float*)d_in[21]};
    const float* lw1 = (const float*)d_in[22];
    const float* lb1 = (const float*)d_in[23];
    const float* lw2 = (const float*)d_in[24];
    const float* lb2 = (const float*)d_in[25];
    float* out = (float*)d_out;

    // workspace layout (floats)
    float* ws = (float*)d_ws;
    size_t off = 0;
    float* HTMP  = ws + off; off += (size_t)NNODE * HC;
    float* H1    = ws + off; off += (size_t)NNODE * HC;
    float* H2    = ws + off; off += (size_t)NNODE * HC;
    float* H3    = ws + off; off += (size_t)NNODE * HC;
    float* AEALL = ws + off; off += (size_t)3 * NEDGE * NHEAD;
    float* ASN   = ws + off; off += (size_t)NNODE * NHEAD;
    float* ADN   = ws + off; off += (size_t)NNODE * NHEAD;
    float* ALPHA = ws + off; off += (size_t)NEDGE * NHEAD;   // doubles as ex
    unsigned* AMAX = (unsigned*)(ws + off); off += (size_t)NNODE * NHEAD;
    float* DENOM = ws + off; off += (size_t)NNODE * NHEAD;
    float* HPOOL = ws + off; off += (size_t)NGRAPH * HC;
    float* ME    = ws + off; off += 256;

    const int TB = 256;                       // 8 waves / block (wave32)
    const int NH  = NNODE * NHEAD;
    const int NHC = NNODE * HC;
    const int EH  = NEDGE * NHEAD;

    // edge logits for all layers (one pass over edge_attr)
    hipLaunchKernelGGL(gat_compute_me, dim3(1), dim3(TB), 0, stream,
                       WE[0], AE[0], WE[1], AE[1], WE[2], AE[2], ME);
    hipLaunchKernelGGL(gat_edge_feat_alpha, dim3((NEDGE + TB - 1) / TB), dim3(TB), 0, stream,
                       edge_attr, ME, AEALL, NEDGE);

    float* HOUT[3] = {H1, H2, H3};
    const float* inptr = x;
    int K = IN_DIM;
    for (int l = 0; l < 3; ++l) {
        int tiles = (NNODE / 16) * (HC / 16);
        hipLaunchKernelGGL(gat_gemm_wmma_f32, dim3((tiles + 7) / 8), dim3(TB), 0, stream,
                           inptr, W[l], HTMP, NNODE, K, HC);
        hipLaunchKernelGGL(gat_node_alpha, dim3((NH + TB - 1) / TB), dim3(TB), 0, stream,
                           HTMP, AS[l], AD[l], ASN, ADN, NH);
        hipLaunchKernelGGL(gat_init_layer, dim3((NHC + TB - 1) / TB), dim3(TB), 0, stream,
                           HOUT[l], DENOM, AMAX, NHC, NH);
        hipLaunchKernelGGL(gat_edge_alpha, dim3((EH + TB - 1) / TB), dim3(TB), 0, stream,
                           ASN, ADN, AEALL + (size_t)l * EH, src, dst, ALPHA, AMAX, EH);
        hipLaunchKernelGGL(gat_edge_exp, dim3((EH + TB - 1) / TB), dim3(TB), 0, stream,
                           ALPHA, AMAX, dst, DENOM, EH);
        hipLaunchKernelGGL(gat_edge_aggregate, dim3((EH + 7) / 8), dim3(TB), 0, stream,
                           HTMP, ALPHA, DENOM, src, dst, HOUT[l], EH);
        hipLaunchKernelGGL(gat_add_bias, dim3((NHC + TB - 1) / TB), dim3(TB), 0, stream,
                           HOUT[l], B[l], NHC);
        inptr = HOUT[l];
        K = HC;
    }

    // pooling + fused MLP head
    int GHC = NGRAPH * HC;
    int initN = (GHC > NNODE) ? GHC : NNODE;
    hipLaunchKernelGGL(gat_init_final, dim3((initN + TB - 1) / TB), dim3(TB), 0, stream,
                       HPOOL, out, lb2, GHC, NNODE);
    hipLaunchKernelGGL(gat_pool, dim3((NHC + TB - 1) / TB), dim3(TB), 0, stream,
                       H3, batch, HPOOL, NHC);
    int tiles2 = (NNODE / 16) * (4 * HC / 16);
    hipLaunchKernelGGL(gat_mlp_fused_wmma, dim3((tiles2 + 7) / 8), dim3(TB), 0, stream,
                       H1, H2, H3, HPOOL, batch, lw1, lb1, lw2, out, NNODE);
}